// InverseWarp_37598143709772
// MI455X (gfx1250) — compile-verified
//
#include <hip/hip_runtime.h>
#include <stdint.h>

// ---------------------------------------------------------------------------
// Problem constants (reference: B=8, src 256x256, dst 512x512, niter=5, ks=3)
// ---------------------------------------------------------------------------
#define BN   8
#define HH   512
#define WW   512
#define HW   (512*512)          // 2^18
#define DOM  524                // padded-reference array side (512 + 2*(niter+1))
#define PW   544                // physical row stride (DOM + guard ring, /16)
#define PP   (544*544)          // per-batch padded array elements
#define NITER 5

typedef float v2f  __attribute__((ext_vector_type(2)));
typedef float v8f  __attribute__((ext_vector_type(8)));
typedef unsigned int u32x4 __attribute__((ext_vector_type(4)));
typedef int   i32x4 __attribute__((ext_vector_type(4)));
typedef int   i32x8 __attribute__((ext_vector_type(8)));

// ---------------------------------------------------------------------------
// TDM: async 18x16 f32 tile load (global -> LDS), descriptor per ISA ch.8.
// group0: count=1 | lds_addr | global_addr(57b) | type=2
// group1: data_size=4B, tensor_dim0=544, tensor_dim1=4096, tile=16x18,
//         tensor_dim0_stride=544 elements.
// This toolchain exposes the 6-arg builtin form.
// ---------------------------------------------------------------------------
__device__ __forceinline__ void tdm_load_tile18x16(const float* gptr,
                                                   unsigned int lds_byte_off) {
  unsigned long long ga = (unsigned long long)(uintptr_t)gptr;
  u32x4 g0;
  g0[0] = 1u;                                   // count=1, no gather
  g0[1] = lds_byte_off;                         // lds_addr
  g0[2] = (unsigned int)ga;                     // global_addr[31:0]
  g0[3] = (unsigned int)((ga >> 32) & 0x01FFFFFFu) | 0x80000000u; // addr[56:32] | type=2
  i32x8 g1;
  g1[0] = (int)(2u << 16);                      // data_size = 2 -> 4 bytes
  g1[1] = (int)(((unsigned)PW & 0xFFFFu) << 16);// tensor_dim0 = 544 (low16)
  g1[2] = (int)(4096u << 16);                   // tensor_dim0 hi=0 | tensor_dim1 lo = 4096
  g1[3] = (int)(16u << 16);                     // tensor_dim1 hi=0 | tile_dim0 = 16
  g1[4] = 18;                                   // tile_dim1 = 18, tile_dim2 = 0
  g1[5] = PW;                                   // tensor_dim0_stride = 544 (low32)
  g1[6] = 0;
  g1[7] = 0;
  i32x4 z4 = {0, 0, 0, 0};
  i32x8 z8 = {0, 0, 0, 0, 0, 0, 0, 0};
  __builtin_amdgcn_tensor_load_to_lds(g0, g1, z4, z4, z8, 0);
}

// Banded Gaussian weight: w(d) = gn[d] for d in {0,1,2}, else 0 (branchless)
__device__ __forceinline__ float wsel(int d, float g0, float g1, float g2) {
  float r = 0.0f;
  r = (d == 0) ? g0 : r;
  r = (d == 1) ? g1 : r;
  r = (d == 2) ? g2 : r;
  return r;
}

// ---------------------------------------------------------------------------
// Separable 3x3 conv of an 18x16 LDS tile via V_WMMA_F32_16X16X4_F32.
// Pass 1 (vertical):   Y(16x16) = Kv(16x20 banded) * X(20x16)  -> 5 WMMAs
// transpose Y through LDS (D-layout -> B-layout)
// Pass 2 (horizontal): Z^T(16x16) = Kh(16x16 banded) * Y^T     -> 4 WMMAs
// Output D-layout: element r of v8f, lane l: col n = r + (l>=16)*8, row m = l&15.
// Valid output cols n in [1,14].
// ---------------------------------------------------------------------------
__device__ __forceinline__ v8f conv3x3_wmma(const volatile float* X, float* ytr,
                                            int lane, float g0, float g1, float g2) {
  const int half = lane >> 4;       // 0: lanes 0-15, 1: lanes 16-31
  const int kkb  = half << 1;       // A/B K offset for this lane half
  const int ml   = lane & 15;
  v8f acc = {0.f, 0.f, 0.f, 0.f, 0.f, 0.f, 0.f, 0.f};
#pragma unroll
  for (int c = 0; c < 5; ++c) {
    int k0 = 4 * c + kkb;
    v2f a, bb;
    a[0]  = wsel(k0 - ml,     g0, g1, g2);      // Kv[m, k] = gn[k-m]
    a[1]  = wsel(k0 + 1 - ml, g0, g1, g2);
    bb[0] = X[k0 * 16 + ml];                    // X row k, col ml
    bb[1] = X[(k0 + 1) * 16 + ml];
    acc = __builtin_amdgcn_wmma_f32_16x16x4_f32(false, a, false, bb,
                                                (short)0, acc, false, false);
  }
  // D-layout -> LDS (row m, col n), stride 17 to dodge bank conflicts
#pragma unroll
  for (int r = 0; r < 8; ++r) ytr[(r + half * 8) * 17 + ml] = acc[r];
  __builtin_amdgcn_wave_barrier();
  v8f acc2 = {0.f, 0.f, 0.f, 0.f, 0.f, 0.f, 0.f, 0.f};
#pragma unroll
  for (int c = 0; c < 4; ++c) {
    int k0 = 4 * c + kkb;
    v2f a, bb;
    a[0]  = wsel(k0 - ml + 1, g0, g1, g2);      // Kh[n, j] = gn[j-n+1]
    a[1]  = wsel(k0 - ml + 2, g0, g1, g2);
    bb[0] = ytr[ml * 17 + k0];                  // Y^T row k = Y col k
    bb[1] = ytr[ml * 17 + k0 + 1];
    acc2 = __builtin_amdgcn_wmma_f32_16x16x4_f32(false, a, false, bb,
                                                 (short)0, acc2, false, false);
  }
  return acc2;
}

// ---------------------------------------------------------------------------
// Kernels
// ---------------------------------------------------------------------------
__global__ void zero_buf(float* __restrict__ p, int n) {
  int i = blockIdx.x * blockDim.x + threadIdx.x;
  if (i < n) p[i] = 0.0f;
}

__global__ void init_winner(int* __restrict__ w, int n) {
  int i = blockIdx.x * blockDim.x + threadIdx.x;
  if (i < n) w[i] = 0x7FFFFFFF;
}

__device__ __forceinline__ float dsrc_at(const float* sb, const float* base,
                                         int sy, int sx, int c) {
  int o = (sy * 256 + sx) * 2 + c;
  return sb[o] - base[o];
}

// Bilinear upsample (src_grid - base) 256^2 -> 512^2, scale to pixel units.
__global__ void resize_disp(const float* __restrict__ src, const float* __restrict__ base,
                            float* __restrict__ dxb, float* __restrict__ dyb, int n) {
  int i = blockIdx.x * blockDim.x + threadIdx.x;
  if (i >= n) return;
  int x = i & (WW - 1), y = (i >> 9) & (HH - 1), b = i >> 18;
  float cy = fminf(fmaxf(0.5f * (float)y - 0.25f, 0.0f), 255.0f);
  float cx = fminf(fmaxf(0.5f * (float)x - 0.25f, 0.0f), 255.0f);
  int y0 = (int)cy, x0 = (int)cx;
  int y1 = min(y0 + 1, 255), x1 = min(x0 + 1, 255);
  float wy = cy - (float)y0, wx = cx - (float)x0;
  const float* sb = src + (size_t)b * 256 * 256 * 2;
  float a0 = dsrc_at(sb, base, y0, x0, 0) * (1.f - wy) + dsrc_at(sb, base, y1, x0, 0) * wy;
  float a1 = dsrc_at(sb, base, y0, x1, 0) * (1.f - wy) + dsrc_at(sb, base, y1, x1, 0) * wy;
  float b0 = dsrc_at(sb, base, y0, x0, 1) * (1.f - wy) + dsrc_at(sb, base, y1, x0, 1) * wy;
  float b1 = dsrc_at(sb, base, y0, x1, 1) * (1.f - wy) + dsrc_at(sb, base, y1, x1, 1) * wy;
  dxb[i] = (a0 * (1.f - wx) + a1 * wx) * (0.5f * WW);
  dyb[i] = (b0 * (1.f - wx) + b1 * wx) * (0.5f * HH);
}

// Dedup pass 1: smallest source index wins per target cell (== stable argsort keep)
__global__ void scatter_min(const float* __restrict__ dxb, const float* __restrict__ dyb,
                            int* __restrict__ winner, int n) {
  int i = blockIdx.x * blockDim.x + threadIdx.x;
  if (i >= n) return;
  int li = i & (HW - 1), b = i >> 18;
  int x = li & 511, y = li >> 9;
  int xi = __float2int_rn((float)x + dxb[i]);   // round-half-even == jnp.round
  int yi = __float2int_rn((float)y + dyb[i]);
  if (xi < 0 || yi < 0 || xi > WW - 1 || yi > HH - 1) return;
  atomicMin(winner + (size_t)b * HW + yi * WW + xi, li);
}

// Dedup pass 2: winners write -dx,-dy,mask into padded arrays (interior offset +8)
__global__ void scatter_write(const float* __restrict__ dxb, const float* __restrict__ dyb,
                              const int* __restrict__ winner,
                              float* __restrict__ idx0, float* __restrict__ idy0,
                              float* __restrict__ m0, int n) {
  int i = blockIdx.x * blockDim.x + threadIdx.x;
  if (i >= n) return;
  int li = i & (HW - 1), b = i >> 18;
  int x = li & 511, y = li >> 9;
  float dx = dxb[i], dy = dyb[i];
  int xi = __float2int_rn((float)x + dx);
  int yi = __float2int_rn((float)y + dy);
  if (xi < 0 || yi < 0 || xi > WW - 1 || yi > HH - 1) return;
  if (winner[(size_t)b * HW + yi * WW + xi] != li) return;
  size_t a = (size_t)b * PP + (size_t)(yi + 8) * PW + (size_t)(xi + 8);
  idx0[a] = -dx;
  idy0[a] = -dy;
  m0[a]   = 1.0f;
}

// One hole-fill iteration over the 524x524 domain.
// 1 wave per tile; TDM stages three 18x16 halo tiles; conv via f32 WMMA.
__global__ __launch_bounds__(32) void fill_step(
    const float* __restrict__ sdx, const float* __restrict__ sdy,
    const float* __restrict__ sm,
    float* __restrict__ ddx, float* __restrict__ ddy, float* __restrict__ dm,
    const float* __restrict__ k2) {
  __shared__ float xdx[20 * 16];
  __shared__ float xdy[20 * 16];
  __shared__ float xms[20 * 16];
  __shared__ float ytr[16 * 17];

  const int lane = threadIdx.x;
  const int u = blockIdx.x, t = blockIdx.y, b = blockIdx.z;
  const int r0 = 16 * t - 1;        // padded-ref row of halo start
  const int c0 = 14 * u - 1;        // padded-ref col of halo start
  const size_t abase = (size_t)b * PP + (size_t)(r0 + 2) * PW + (size_t)(c0 + 2);

  // zero the K-padding rows 18..19 (read by the K=20 WMMA chunk, weight 0)
  xdx[288 + lane] = 0.0f;
  xdy[288 + lane] = 0.0f;
  xms[288 + lane] = 0.0f;
  __builtin_amdgcn_wave_barrier();

  tdm_load_tile18x16(sdx + abase, (unsigned int)(uintptr_t)(&xdx[0]));
  tdm_load_tile18x16(sdy + abase, (unsigned int)(uintptr_t)(&xdy[0]));
  tdm_load_tile18x16(sm  + abase, (unsigned int)(uintptr_t)(&xms[0]));
  __builtin_amdgcn_s_wait_tensorcnt(0);
  __asm__ volatile("" ::: "memory");

  // separable weights: row-sums of the (normalized) 3x3 kernel
  float g0 = k2[0] + k2[1] + k2[2];
  float g1 = k2[3] + k2[4] + k2[5];
  float g2 = k2[6] + k2[7] + k2[8];

  volatile float* vdx = xdx;
  volatile float* vdy = xdy;
  volatile float* vms = xms;

  v8f zdx = conv3x3_wmma(vdx, ytr, lane, g0, g1, g2);
  __builtin_amdgcn_wave_barrier();
  v8f zdy = conv3x3_wmma(vdy, ytr, lane, g0, g1, g2);
  __builtin_amdgcn_wave_barrier();
  v8f zms = conv3x3_wmma(vms, ytr, lane, g0, g1, g2);

  const int half = lane >> 4, ml = lane & 15;
  const int py = r0 + 1 + ml;       // output padded-ref row for this lane
#pragma unroll
  for (int r = 0; r < 8; ++r) {
    int n = r + half * 8;           // tile-local output col
    if (n < 1 || n > 14) continue;  // (divergence only after all WMMAs)
    int px = c0 + n;
    if (py >= DOM || px >= DOM) continue;
    float mo = vms[(ml + 1) * 16 + n];
    bool old = mo > 0.5f;
    bool nb  = (vms[ml * 16 + n] > 0.5f) || (vms[(ml + 2) * 16 + n] > 0.5f) ||
               (vms[(ml + 1) * 16 + n - 1] > 0.5f) || (vms[(ml + 1) * 16 + n + 1] > 0.5f);
    bool nm = (!old) && nb;
    float denom = nm ? zms[r] : 1.0f;
    float odx = nm ? zdx[r] / denom : vdx[(ml + 1) * 16 + n];
    float ody = nm ? zdy[r] / denom : vdy[(ml + 1) * 16 + n];
    float om  = (old || nm) ? 1.0f : 0.0f;
    size_t a = (size_t)b * PP + (size_t)(py + 2) * PW + (size_t)(px + 2);
    ddx[a] = odx;
    ddy[a] = ody;
    dm[a]  = om;
  }
}

// One erosion iteration (True padding beyond the 524 domain boundary)
__device__ __forceinline__ bool mask_at(const float* p, int yy, int xx) {
  if (yy < 0 || yy >= DOM || xx < 0 || xx >= DOM) return true;
  return p[(size_t)(yy + 2) * PW + (xx + 2)] > 0.5f;
}

__global__ void erode_step(const float* __restrict__ ms, float* __restrict__ md, int n) {
  int i = blockIdx.x * blockDim.x + threadIdx.x;
  if (i >= n) return;
  int x = i % DOM;
  int tt = i / DOM;
  int y = tt % DOM;
  int b = tt / DOM;
  const float* p = ms + (size_t)b * PP;
  bool c = p[(size_t)(y + 2) * PW + (x + 2)] > 0.5f;
  bool v = c && mask_at(p, y - 1, x) && mask_at(p, y + 1, x) &&
               mask_at(p, y, x - 1) && mask_at(p, y, x + 1);
  md[(size_t)b * PP + (size_t)(y + 2) * PW + (x + 2)] = v ? 1.0f : 0.0f;
}

// out = tgt_grid + stack(inv_dx * 2/W, inv_dy * 2/H), masked-out -> 2W / 2H
__global__ void compose_out(const float* __restrict__ fdx, const float* __restrict__ fdy,
                            const float* __restrict__ fm, const float* __restrict__ tgt,
                            float* __restrict__ out, int n) {
  int i = blockIdx.x * blockDim.x + threadIdx.x;
  if (i >= n) return;
  int x = i & 511, y = (i >> 9) & 511, b = i >> 18;
  size_t a = (size_t)b * PP + (size_t)(y + 8) * PW + (size_t)(x + 8);
  bool m = fm[a] > 0.5f;
  float vx = m ? fdx[a] : 2.0f * WW;
  float vy = m ? fdy[a] : 2.0f * HH;
  int to = (y * WW + x) * 2;
  out[(size_t)b * HW * 2 + to]     = tgt[to]     + vx * (2.0f / WW);
  out[(size_t)b * HW * 2 + to + 1] = tgt[to + 1] + vy * (2.0f / HH);
}

// ---------------------------------------------------------------------------
// Host launcher
// ---------------------------------------------------------------------------
extern "C" void kernel_launch(void* const* d_in, const int* in_sizes, int n_in,
                              void* d_out, int out_size, void* d_ws, size_t ws_size,
                              hipStream_t stream) {
  (void)in_sizes; (void)n_in; (void)out_size; (void)ws_size;
  const float* src_grid = (const float*)d_in[0];
  const float* kern     = (const float*)d_in[1];
  const float* src_base = (const float*)d_in[2];
  const float* tgt_grid = (const float*)d_in[3];
  float* out = (float*)d_out;

  const size_t N1 = (size_t)BN * HW;            // 2,097,152
  const size_t NF = (size_t)BN * PP;            // 2,367,488
  float* dxb    = (float*)d_ws;
  float* dyb    = dxb + N1;
  int*   winner = (int*)(dyb + N1);
  float* f      = (float*)(winner + N1);
  float* Adx[2] = { f,          f + NF     };
  float* Ady[2] = { f + 2 * NF, f + 3 * NF };
  float* Am[2]  = { f + 4 * NF, f + 5 * NF };

  const int T = 256;
  zero_buf<<<(int)((6 * NF + T - 1) / T), T, 0, stream>>>(f, (int)(6 * NF));
  init_winner<<<(int)((N1 + T - 1) / T), T, 0, stream>>>(winner, (int)N1);
  resize_disp<<<(int)((N1 + T - 1) / T), T, 0, stream>>>(src_grid, src_base, dxb, dyb, (int)N1);
  scatter_min<<<(int)((N1 + T - 1) / T), T, 0, stream>>>(dxb, dyb, winner, (int)N1);
  scatter_write<<<(int)((N1 + T - 1) / T), T, 0, stream>>>(dxb, dyb, winner,
                                                           Adx[0], Ady[0], Am[0], (int)N1);

  dim3 fgrid(38, 33, BN);                       // 14-col x 16-row output tiles
  int cur = 0;
  for (int it = 0; it < NITER; ++it) {
    fill_step<<<fgrid, 32, 0, stream>>>(Adx[cur], Ady[cur], Am[cur],
                                        Adx[1 - cur], Ady[1 - cur], Am[1 - cur], kern);
    cur = 1 - cur;                              // ends at 1
  }

  const size_t ND = (size_t)BN * DOM * DOM;
  int mcur = cur;
  for (int it = 0; it < NITER; ++it) {
    erode_step<<<(int)((ND + T - 1) / T), T, 0, stream>>>(Am[mcur], Am[1 - mcur], (int)ND);
    mcur = 1 - mcur;                            // ends at 0
  }

  compose_out<<<(int)((N1 + T - 1) / T), T, 0, stream>>>(Adx[cur], Ady[cur], Am[mcur],
                                                         tgt_grid, out, (int)N1);
}